// CRPSLoss_17386027614770
// MI455X (gfx1250) — compile-verified
//
#include <hip/hip_runtime.h>

#define N_ROWS 4096
#define M_COLS 256

typedef __attribute__((ext_vector_type(2))) float v2f;
typedef __attribute__((ext_vector_type(8))) float v8f;

// Kernel A: one block per row. Stage row in LDS, brute-force all-pairs
// |x_k - x_l| (broadcast LDS reads, float4-vectorized), plus |x_k - y_i|.
// Block-reduce both and emit two partials per row into workspace.
__global__ __launch_bounds__(256) void crps_row_kernel(
    const float* __restrict__ y_pred, const float* __restrict__ y,
    float* __restrict__ ws_pair, float* __restrict__ ws_mae) {
  __shared__ __align__(16) float sm[M_COLS];
  const int i = blockIdx.x;
  const int k = threadIdx.x;

  const float x  = y_pred[i * M_COLS + k];
  const float yv = y[i];
  sm[k] = x;
  __syncthreads();

  const float mae = fabsf(x - yv);

  float p = 0.0f;
  const float4* r4 = (const float4*)sm;
#pragma unroll 4
  for (int l = 0; l < M_COLS / 4; ++l) {
    float4 v = r4[l];  // broadcast ds_load_b128 (all lanes same address)
    p += fabsf(x - v.x);
    p += fabsf(x - v.y);
    p += fabsf(x - v.z);
    p += fabsf(x - v.w);
  }

  // --- block tree-reduce p ---
  __syncthreads();
  sm[k] = p;
  __syncthreads();
#pragma unroll
  for (int s = 128; s > 0; s >>= 1) {
    if (k < s) sm[k] += sm[k + s];
    __syncthreads();
  }
  if (k == 0) ws_pair[i] = sm[0];
  __syncthreads();

  // --- block tree-reduce mae ---
  sm[k] = mae;
  __syncthreads();
#pragma unroll
  for (int s = 128; s > 0; s >>= 1) {
    if (k < s) sm[k] += sm[k + s];
    __syncthreads();
  }
  if (k == 0) ws_mae[i] = sm[0];
}

// Kernel B: single wave (32 lanes, EXEC all-ones as WMMA requires).
// Reduce the 2x4096 partials with chained V_WMMA_F32_16X16X4_F32:
// B = ones  =>  D[m][n] += rowsum(A[m][:]) , so each WMMA folds a 64-float
// tile into 16 fp32 accumulator chains. 64 WMMAs per array.
__global__ __launch_bounds__(32) void crps_final_kernel(
    const float* __restrict__ ws_pair, const float* __restrict__ ws_mae,
    float* __restrict__ out) {
  const int lane = threadIdx.x;
  const int m    = lane & 15;         // A-matrix row for this lane
  const int kb   = (lane >> 4) * 2;   // K base: lanes 0-15 -> K{0,1}, 16-31 -> K{2,3}

  v2f bones;
  bones.x = 1.0f;
  bones.y = 1.0f;

  v8f cp = {};
  v8f cm = {};

  for (int t = 0; t < N_ROWS / 64; ++t) {
    const float* base = ws_pair + t * 64 + m * 4 + kb;
    v2f a;
    a.x = base[0];
    a.y = base[1];
    cp = __builtin_amdgcn_wmma_f32_16x16x4_f32(
        /*neg_a=*/false, a, /*neg_b=*/false, bones,
        /*c_mod=*/(short)0, cp, /*reuse_a=*/false, /*reuse_b=*/false);
  }
  for (int t = 0; t < N_ROWS / 64; ++t) {
    const float* base = ws_mae + t * 64 + m * 4 + kb;
    v2f a;
    a.x = base[0];
    a.y = base[1];
    cm = __builtin_amdgcn_wmma_f32_16x16x4_f32(
        false, a, false, bones, (short)0, cm, false, false);
  }

  // Per-lane: sum the 8 accumulator VGPRs. Lane 0 holds M=0..7 of column 0,
  // lane 16 holds M=8..15 of column 0; columns are identical since B = ones.
  float sp = cp[0] + cp[1] + cp[2] + cp[3] + cp[4] + cp[5] + cp[6] + cp[7];
  float sa = cm[0] + cm[1] + cm[2] + cm[3] + cm[4] + cm[5] + cm[6] + cm[7];

  float pair_total = __shfl(sp, 0, 32) + __shfl(sp, 16, 32);
  float mae_total  = __shfl(sa, 0, 32) + __shfl(sa, 16, 32);

  if (lane == 0) {
    const float inv_nm   = 1.0f / ((float)N_ROWS * (float)M_COLS);          // 1/1048576
    const float inv_pair = 1.0f / (2.0f * (float)N_ROWS * (float)M_COLS * (float)M_COLS); // 1/536870912
    out[0] = mae_total * inv_nm - pair_total * inv_pair;
  }
}

extern "C" void kernel_launch(void* const* d_in, const int* in_sizes, int n_in,
                              void* d_out, int out_size, void* d_ws, size_t ws_size,
                              hipStream_t stream) {
  const float* y_pred = (const float*)d_in[0];
  const float* y      = (const float*)d_in[1];
  float* ws_pair = (float*)d_ws;            // 4096 floats
  float* ws_mae  = ws_pair + N_ROWS;        // 4096 floats (32 KB total)

  crps_row_kernel<<<N_ROWS, 256, 0, stream>>>(y_pred, y, ws_pair, ws_mae);
  crps_final_kernel<<<1, 32, 0, stream>>>(ws_pair, ws_mae, (float*)d_out);
}